// DecodeTBPPPredictions_8564164789047
// MI455X (gfx1250) — compile-verified
//
#include <hip/hip_runtime.h>
#include <hip/hip_bf16.h>

#define BATCH    16
#define NBOX     120000
#define KTOP     400
#define NPRED    10
#define CANDCAP  512
#define TROWS    6000
#define CHUNKS   (NBOX / TROWS)       // 20
#define CONF_THR 0.01f
#define IOU_THR  0.45f
#define INPUT_SZ 384.0f

typedef unsigned int v4u __attribute__((ext_vector_type(4)));
typedef int          v8i __attribute__((ext_vector_type(8)));
typedef int          v4i __attribute__((ext_vector_type(4)));

// ---------- monotonic float <-> uint order mapping ----------
__device__ __forceinline__ unsigned int f_ord(float f) {
    unsigned int u = __float_as_uint(f);
    return (u & 0x80000000u) ? ~u : (u | 0x80000000u);
}
__device__ __forceinline__ float f_unord(unsigned int o) {
    unsigned int u = (o & 0x80000000u) ? (o & 0x7FFFFFFFu) : ~o;
    return __uint_as_float(u);
}

// ---------- Tensor Data Mover: 2-D tile (4-byte elements) global -> LDS ----------
// D# packing per CDNA5 ISA sec 8.3/8.4.  clang-23 6-arg builtin form:
// (uint32x4 g0, int32x8 g1, int32x4 g2, int32x4 g3, int32x8 gx, i32 cpol)
__device__ __forceinline__ void tdm_load_2d(unsigned int lds_off,
                                            unsigned long long gaddr,
                                            unsigned int tensor_d0,
                                            unsigned int tensor_d1,
                                            unsigned long long stride0,
                                            unsigned int tile_d0,
                                            unsigned int tile_d1) {
    v4u g0;
    g0.x = 1u;                                         // count=1 (valid), user mode
    g0.y = lds_off;                                    // lds_addr (bytes)
    g0.z = (unsigned int)(gaddr & 0xFFFFFFFFull);      // global_addr[31:0]
    g0.w = (unsigned int)((gaddr >> 32) & 0x1FFFFFFull)// global_addr[56:32]
           | 0x80000000u;                              // type=2 ("image") at [127:126]
    v8i g1;
    g1[0] = (int)(2u << 16);                           // data_size=2 -> 4 bytes; no mcast/pad
    g1[1] = (int)((tensor_d0 & 0xFFFFu) << 16);        // tensor_dim0[15:0] at bits 63:48
    g1[2] = (int)((tensor_d0 >> 16) | ((tensor_d1 & 0xFFFFu) << 16)); // d0 hi | d1 lo
    g1[3] = (int)((tensor_d1 >> 16) | ((tile_d0 & 0xFFFFu) << 16));   // d1 hi | tile_dim0
    g1[4] = (int)(tile_d1 & 0xFFFFu);                  // tile_dim1 | tile_dim2=0
    g1[5] = (int)(stride0 & 0xFFFFFFFFull);            // tensor_dim0_stride lo
    g1[6] = (int)((stride0 >> 32) & 0xFFFFull);        // stride0 hi | dim1_stride lo = 0
    g1[7] = 0;
    v4i zz4 = {0, 0, 0, 0};
    v8i zz8 = {0, 0, 0, 0, 0, 0, 0, 0};
    __builtin_amdgcn_tensor_load_to_lds(g0, g1, zz4, zz4, zz8, 0);
}

// ---------- kernel 0: zero per-batch candidate counters ----------
__global__ void k_init(unsigned int* cand_cnt) {
    int t = threadIdx.x;
    if (t < BATCH) cand_cnt[t] = 0u;
}

// ---------- kernel 1: TDM strided gather of score column -> contiguous ws ----------
__global__ void k_gather(const float* __restrict__ y, float* __restrict__ scores) {
    __shared__ float sm[TROWS];                        // 24 KB
    const int blk = blockIdx.x;
    const int b   = blk / CHUNKS;
    const int ch  = blk - b * CHUNKS;
    const size_t r0 = (size_t)ch * TROWS;
    if (threadIdx.x < 32) {                            // one wave issues the DMA
        unsigned long long ga =
            (unsigned long long)(const void*)(y + ((size_t)b * NBOX + r0) * 22 + 1);
        tdm_load_2d((unsigned int)(unsigned long long)(void*)sm, ga,
                    /*tensor_d0=*/22, /*tensor_d1=*/TROWS,
                    /*stride0=*/22ull, /*tile_d0=*/1, /*tile_d1=*/TROWS);
        __builtin_amdgcn_s_wait_tensorcnt(0);
    }
    __syncthreads();
    for (int i = threadIdx.x; i < TROWS; i += blockDim.x)
        scores[(size_t)b * NBOX + r0 + i] = sm[i];
}

// ---------- kernel 2: per-batch 4-pass radix select of the 400th-largest ----------
__global__ void k_radix(const float* __restrict__ scores, unsigned int* __restrict__ thr) {
    __shared__ unsigned int hist[256];
    __shared__ unsigned int sprefix, srank;
    const int b = blockIdx.x, tid = threadIdx.x;
    if (tid == 0) { sprefix = 0u; srank = KTOP; }
    for (int p = 3; p >= 0; --p) {
        for (int i = tid; i < 256; i += blockDim.x) hist[i] = 0u;
        __syncthreads();
        const unsigned int prefix = sprefix;
        const unsigned int shift  = (unsigned int)p * 8u;
        const unsigned int mhi    = (p == 3) ? 0u : (0xFFFFFFFFu << (shift + 8u));
        for (int i = tid; i < NBOX; i += blockDim.x) {
            unsigned int o = f_ord(scores[(size_t)b * NBOX + i]);
            if ((o & mhi) == prefix) atomicAdd(&hist[(o >> shift) & 0xFFu], 1u);
        }
        __syncthreads();
        if (tid == 0) {
            unsigned int r = srank, acc = 0u;
            int bin = 255;
            for (; bin > 0; --bin) {
                unsigned int c = hist[bin];
                if (acc + c >= r) break;
                acc += c;
            }
            srank   = r - acc;
            sprefix = prefix | ((unsigned int)bin << shift);
        }
        __syncthreads();
    }
    if (tid == 0) thr[b] = sprefix;
}

// ---------- kernel 3: compact every score >= threshold into (ord<<32 | ~idx) keys ----------
__global__ void k_compact(const float* __restrict__ scores,
                          const unsigned int* __restrict__ thr,
                          unsigned int* __restrict__ cand_cnt,
                          unsigned long long* __restrict__ keys) {
    int g = blockIdx.x * blockDim.x + threadIdx.x;
    if (g >= BATCH * NBOX) return;
    int b = g / NBOX;
    int i = g - b * NBOX;
    unsigned int o = f_ord(scores[g]);
    if (o >= thr[b]) {
        unsigned int pos = atomicAdd(&cand_cnt[b], 1u);
        if (pos < CANDCAP)
            keys[(size_t)b * CANDCAP + pos] =
                ((unsigned long long)o << 32) | (unsigned int)(~(unsigned int)i);
    }
}

// ---------- kernel 4: per-batch bitonic sort (desc score, asc index) -> top-400 ----------
__global__ void k_sort(const unsigned long long* __restrict__ keys,
                       const unsigned int* __restrict__ cand_cnt,
                       float* __restrict__ ts, unsigned int* __restrict__ ti) {
    __shared__ unsigned long long sk[CANDCAP];
    const int b = blockIdx.x, tid = threadIdx.x;
    unsigned int n = cand_cnt[b];
    if (n > CANDCAP) n = CANDCAP;
    sk[tid] = (tid < (int)n) ? keys[(size_t)b * CANDCAP + tid] : 0ull;
    __syncthreads();
    for (int k = 2; k <= CANDCAP; k <<= 1) {
        for (int j = k >> 1; j > 0; j >>= 1) {
            int ixj = tid ^ j;
            if (ixj > tid) {
                bool desc = ((tid & k) == 0);
                unsigned long long a = sk[tid], c = sk[ixj];
                bool sw = desc ? (a < c) : (a > c);
                if (sw) { sk[tid] = c; sk[ixj] = a; }
            }
            __syncthreads();
        }
    }
    if (tid < KTOP) {
        unsigned long long key = sk[tid];
        ts[b * KTOP + tid] = f_unord((unsigned int)(key >> 32));
        ti[b * KTOP + tid] = ~((unsigned int)key);
    }
}

// ---------- kernel 5: decode only the 16x400 selected anchors ----------
__global__ void k_decode(const float* __restrict__ y,
                         const unsigned int* __restrict__ ti,
                         float* __restrict__ boxes, float* __restrict__ quads) {
    int g = blockIdx.x * blockDim.x + threadIdx.x;
    if (g >= BATCH * KTOP) return;
    int b = g / KTOP;
    size_t base = ((size_t)b * NBOX + ti[g]) * 22;
    float v[22];
#pragma unroll
    for (int c = 0; c < 22; ++c) v[c] = y[base + c];
    // loc = v[2..13], db = v[14..17] (dcx,dcy,dw,dh), var = v[18..21]
    const float dcx = v[14], dcy = v[15], dw = v[16], dh = v[17];
    const float v0 = v[18], v1 = v[19], v2 = v[20], v3 = v[21];
    const float cx = v[2] * v0 * dw + dcx;
    const float cy = v[3] * v1 * dh + dcy;
    const float w  = expf(v[4] * v2) * dw;
    const float h  = expf(v[5] * v3) * dh;
    boxes[(size_t)g * 4 + 0] = (cx - 0.5f * w) * INPUT_SZ;
    boxes[(size_t)g * 4 + 1] = (cy - 0.5f * h) * INPUT_SZ;
    boxes[(size_t)g * 4 + 2] = (cx + 0.5f * w) * INPUT_SZ;
    boxes[(size_t)g * 4 + 3] = (cy + 0.5f * h) * INPUT_SZ;
    const float dqx[4] = { dcx - 0.5f * dw, dcx + 0.5f * dw, dcx + 0.5f * dw, dcx - 0.5f * dw };
    const float dqy[4] = { dcy - 0.5f * dh, dcy - 0.5f * dh, dcy + 0.5f * dh, dcy + 0.5f * dh };
#pragma unroll
    for (int m = 0; m < 4; ++m) {
        quads[(size_t)g * 8 + 2 * m + 0] = (dqx[m] + v[6 + 2 * m] * v0 * dw) * INPUT_SZ;
        quads[(size_t)g * 8 + 2 * m + 1] = (dqy[m] + v[7 + 2 * m] * v1 * dh) * INPUT_SZ;
    }
}

// ---------- kernel 6: per-batch NMS (TDM box load, bitmask IoU, wave32 ballot loop) ----------
__global__ void k_nms(const float* __restrict__ ts,
                      const float* __restrict__ boxes,
                      const float* __restrict__ quads,
                      float* __restrict__ out) {
    __shared__ float        sb[KTOP * 4];      // boxes, filled by TDM
    __shared__ float        ss[KTOP];          // top scores
    __shared__ float        sarea[KTOP];
    __shared__ unsigned int smask[KTOP * 16];  // j<i suppression bits (iou > thr)
    __shared__ unsigned int skeep[16];
    __shared__ int          ssel[NPRED];
    __shared__ int          sflag[NPRED];
    const int b = blockIdx.x, tid = threadIdx.x;

    if (tid < 32) {                            // wave 0: DMA 1600 floats of boxes into LDS
        unsigned long long ga =
            (unsigned long long)(const void*)(boxes + (size_t)b * KTOP * 4);
        tdm_load_2d((unsigned int)(unsigned long long)(void*)sb, ga,
                    /*tensor_d0=*/KTOP * 4, /*tensor_d1=*/1,
                    /*stride0=*/(unsigned long long)(KTOP * 4),
                    /*tile_d0=*/KTOP * 4, /*tile_d1=*/1);
        __builtin_amdgcn_s_wait_tensorcnt(0);
    }
    for (int i = tid; i < KTOP; i += blockDim.x) ss[i] = ts[b * KTOP + i];
    for (int w = tid; w < KTOP * 16; w += blockDim.x) smask[w] = 0u;
    __syncthreads();

    for (int i = tid; i < KTOP; i += blockDim.x)
        sarea[i] = (sb[i * 4 + 2] - sb[i * 4 + 0]) * (sb[i * 4 + 3] - sb[i * 4 + 1]);
    __syncthreads();

    for (int p = tid; p < KTOP * KTOP; p += blockDim.x) {
        int i = p / KTOP, j = p - i * KTOP;
        if (j >= i) continue;
        float ix1 = fmaxf(sb[i * 4 + 0], sb[j * 4 + 0]);
        float iy1 = fmaxf(sb[i * 4 + 1], sb[j * 4 + 1]);
        float ix2 = fminf(sb[i * 4 + 2], sb[j * 4 + 2]);
        float iy2 = fminf(sb[i * 4 + 3], sb[j * 4 + 3]);
        float iw = fmaxf(ix2 - ix1, 0.0f);
        float ih = fmaxf(iy2 - iy1, 0.0f);
        float inter = iw * ih;
        float uni = sarea[i] + sarea[j] - inter;
        float iou = inter / fmaxf(uni, 1e-8f);
        if (iou > IOU_THR) atomicOr(&smask[i * 16 + (j >> 5)], 1u << (j & 31));
    }
    __syncthreads();

    if (tid < 32) {                            // serial greedy NMS, bitset across wave 0
        const int lane = tid;
        unsigned int keepw = 0u;
        for (int i = 0; i < KTOP; ++i) {
            int wi = i >> 5;
            unsigned int bi = (unsigned int)(i & 31);
            unsigned int lt = (lane < wi) ? 0xFFFFFFFFu
                              : ((lane == wi && bi) ? ((1u << bi) - 1u) : 0u);
            unsigned int row = (lane < 16) ? smask[i * 16 + lane] : 0u;
            unsigned int t = row & keepw & lt;
            unsigned int anyb = __builtin_amdgcn_ballot_w32(t != 0u);
            bool kp = (ss[i] > CONF_THR) && (anyb == 0u);
            if (kp && lane == wi) keepw |= (1u << bi);
        }
        if (lane < 16) skeep[lane] = keepw;
    }
    __syncthreads();

    if (tid == 0) {                            // top-10 of masked scores (ts already desc)
        int cnt = 0;
        for (int i = 0; i < KTOP && cnt < NPRED; ++i)
            if ((skeep[i >> 5] >> (i & 31)) & 1u) { ssel[cnt] = i; sflag[cnt] = 1; ++cnt; }
        for (int i = 0; i < KTOP && cnt < NPRED; ++i)
            if (!((skeep[i >> 5] >> (i & 31)) & 1u)) { ssel[cnt] = i; sflag[cnt] = 0; ++cnt; }
    }
    __syncthreads();

    if (tid < NPRED * 13) {
        int n = tid / 13, c = tid - n * 13;
        int k = ssel[n];
        float val;
        if (c == 0)      val = sflag[n] ? ss[k] : -1.0f;
        else if (c <= 4) val = sb[k * 4 + (c - 1)];
        else             val = quads[((size_t)b * KTOP + k) * 8 + (c - 5)];
        out[(size_t)b * (NPRED * 13) + tid] = val;
    }
}

extern "C" void kernel_launch(void* const* d_in, const int* in_sizes, int n_in,
                              void* d_out, int out_size, void* d_ws, size_t ws_size,
                              hipStream_t stream) {
    (void)in_sizes; (void)n_in; (void)out_size; (void)ws_size;
    const float* y = (const float*)d_in[0];
    float* out = (float*)d_out;
    char* ws = (char*)d_ws;

    // workspace layout (bytes)
    const size_t OFF_SCORES = 0;                                   // 16*120000*4 = 7,680,000
    const size_t OFF_THR    = OFF_SCORES + (size_t)BATCH * NBOX * 4;
    const size_t OFF_CNT    = OFF_THR + 256;
    const size_t OFF_KEYS   = OFF_CNT + 256;                       // 16*512*8 = 65536
    const size_t OFF_TS     = OFF_KEYS + (size_t)BATCH * CANDCAP * 8;
    const size_t OFF_TI     = OFF_TS + (size_t)BATCH * KTOP * 4;
    const size_t OFF_BOX    = OFF_TI + (size_t)BATCH * KTOP * 4;
    const size_t OFF_QUAD   = OFF_BOX + (size_t)BATCH * KTOP * 16;

    float*              scores = (float*)(ws + OFF_SCORES);
    unsigned int*       thr    = (unsigned int*)(ws + OFF_THR);
    unsigned int*       cnt    = (unsigned int*)(ws + OFF_CNT);
    unsigned long long* keys   = (unsigned long long*)(ws + OFF_KEYS);
    float*              ts     = (float*)(ws + OFF_TS);
    unsigned int*       ti     = (unsigned int*)(ws + OFF_TI);
    float*              boxes  = (float*)(ws + OFF_BOX);
    float*              quads  = (float*)(ws + OFF_QUAD);

    k_init   <<<1, 64, 0, stream>>>(cnt);
    k_gather <<<BATCH * CHUNKS, 256, 0, stream>>>(y, scores);
    k_radix  <<<BATCH, 1024, 0, stream>>>(scores, thr);
    k_compact<<<(BATCH * NBOX + 255) / 256, 256, 0, stream>>>(scores, thr, cnt, keys);
    k_sort   <<<BATCH, CANDCAP, 0, stream>>>(keys, cnt, ts, ti);
    k_decode <<<(BATCH * KTOP + 127) / 128, 128, 0, stream>>>(y, ti, boxes, quads);
    k_nms    <<<BATCH, 512, 0, stream>>>(ts, boxes, quads, out);
}